// MoERouter_10986526343381
// MI455X (gfx1250) — compile-verified
//
#include <hip/hip_runtime.h>

typedef __attribute__((ext_vector_type(2))) float        v2f;
typedef __attribute__((ext_vector_type(8))) float        v8f;
typedef __attribute__((ext_vector_type(4))) unsigned int v4u;
typedef __attribute__((ext_vector_type(8))) int          v8i;
typedef __attribute__((ext_vector_type(4))) int          v4i;

#define NUM_EXPERTS 64
#define TOP_K 8
#define HIDDEN 2048
#define NUM_TOKENS 16384

#define WAVES_PER_BLOCK 4
#define TOKENS_PER_WAVE 16
#define TOKENS_PER_BLOCK (WAVES_PER_BLOCK * TOKENS_PER_WAVE)   // 64
#define ROW_STRIDE 65        // logits LDS: 64 + 1 pad -> conflict-free

#define CHUNK_K 64
#define NCHUNK (HIDDEN / CHUNK_K)   // 32
#define B_STRIDE 68          // 64 dw + 4 dw TDM pad -> all-64-bank conflict-free, 16B-aligned rows

__global__ void moe_zero_counts_kernel(float* __restrict__ counts) {
    counts[threadIdx.x] = 0.0f;
}

// One TDM descriptor: 2D tile (64 K-values x 64 experts) of gate_w -> LDS,
// padded +4 dwords per 64-dword row so LDS rows have stride B_STRIDE.
__device__ __forceinline__ void tdm_load_b_chunk(const float* gw_chunk, unsigned lds_bytes) {
    unsigned long long ga = (unsigned long long)(size_t)gw_chunk;
    v4u g0 = { 1u,                                                   // count=1, user descriptor
               lds_bytes,                                            // lds_addr
               (unsigned)ga,                                         // global_addr[31:0]
               ((unsigned)(ga >> 32) & 0x01FFFFFFu) | 0x80000000u }; // global_addr[56:32] | type=2
    v8i g1 = { (int)((2u << 16)    // data_size = 4 bytes
                   | (1u << 20)    // pad_enable
                   | (5u << 22)    // pad_interval: 2^(5+1)=64 dwords between pads
                   | (3u << 25)),  // pad_amount: 3+1 = 4 dwords
               (int)(((unsigned)HIDDEN & 0xFFFFu) << 16),                          // tensor_dim0[15:0]
               (int)((((unsigned)HIDDEN >> 16) & 0xFFFFu) | ((unsigned)NUM_EXPERTS << 16)), // dim0 hi | dim1 lo
               (int)((unsigned)CHUNK_K << 16),                                     // dim1 hi=0 | tile_dim0=64
               (int)(unsigned)NUM_EXPERTS,                                         // tile_dim1=64, tile_dim2=0
               (int)HIDDEN,                                                        // tensor_dim0_stride lo32
               0, 0 };                                                             // stride hi, dim1_stride (2D: unused)
    v4i g2 = { 0, 0, 0, 0 };
    v4i g3 = { 0, 0, 0, 0 };
#if __clang_major__ >= 23
    v8i g4 = { 0, 0, 0, 0, 0, 0, 0, 0 };
    __builtin_amdgcn_tensor_load_to_lds(g0, g1, g2, g3, g4, 0);
#else
    __builtin_amdgcn_tensor_load_to_lds(g0, g1, g2, g3, 0);
#endif
}

__global__ __launch_bounds__(WAVES_PER_BLOCK * 32)
void moe_router_kernel(const float* __restrict__ x,
                       const float* __restrict__ gate_w,
                       const float* __restrict__ expert_bias,
                       int*   __restrict__ out_idx,
                       float* __restrict__ out_scr,
                       float* __restrict__ out_counts) {
    __shared__ float s_b[2][NUM_EXPERTS * B_STRIDE];   // double-buffered gate_w chunk
    __shared__ float s_logits[WAVES_PER_BLOCK * TOKENS_PER_WAVE * ROW_STRIDE];
    __shared__ float s_bias[NUM_EXPERTS];
    __shared__ float s_counts[NUM_EXPERTS];

    const int tid  = threadIdx.x;
    const int wave = tid >> 5;
    const int lane = tid & 31;
    const int half = lane >> 4;   // 0: lanes 0-15 (K=0,1), 1: lanes 16-31 (K=2,3)
    const int lmod = lane & 15;

    if (tid < NUM_EXPERTS) {
        s_bias[tid]   = expert_bias[tid];
        s_counts[tid] = 0.0f;
    }

    const int token_base = blockIdx.x * TOKENS_PER_BLOCK + wave * TOKENS_PER_WAVE;

    // Prologue: wave 0 DMAs chunk 0 of gate_w into buffer 0 and waits.
    if (wave == 0) {
        tdm_load_b_chunk(gate_w, (unsigned)(size_t)&s_b[0][0]);
        __builtin_amdgcn_s_wait_tensorcnt(0);
    }

    // A fragment (16x4 fp32 layout): per-lane float2 from this lane's token row.
    const float* aBase = x + (size_t)(token_base + lmod) * HIDDEN + half * 2;
    // B fragment base (words) inside an LDS chunk buffer for expert tile 0.
    const int bword = lmod * B_STRIDE + half * 2;

    v8f acc0 = {}, acc1 = {}, acc2 = {}, acc3 = {};

    for (int c = 0; c < NCHUNK; ++c) {
        __syncthreads();   // publish chunk c to all waves; prior compute on other buffer done
        if (wave == 0 && (c + 1) < NCHUNK) {
            tdm_load_b_chunk(gate_w + (size_t)(c + 1) * CHUNK_K,
                             (unsigned)(size_t)&s_b[(c + 1) & 1][0]);
        }
        const float* bBuf = &s_b[c & 1][0];
        const float* aPtr = aBase + c * CHUNK_K;

        #pragma unroll
        for (int kk = 0; kk < CHUNK_K; kk += 4) {
            v2f a  = *(const v2f*)(aPtr + kk);
            v2f b0 = *(const v2f*)(bBuf + bword + kk + 0 * 16 * B_STRIDE);
            v2f b1 = *(const v2f*)(bBuf + bword + kk + 1 * 16 * B_STRIDE);
            v2f b2 = *(const v2f*)(bBuf + bword + kk + 2 * 16 * B_STRIDE);
            v2f b3 = *(const v2f*)(bBuf + bword + kk + 3 * 16 * B_STRIDE);
            acc0 = __builtin_amdgcn_wmma_f32_16x16x4_f32(false, a, false, b0, (short)0, acc0, false, false);
            acc1 = __builtin_amdgcn_wmma_f32_16x16x4_f32(false, a, false, b1, (short)0, acc1, false, false);
            acc2 = __builtin_amdgcn_wmma_f32_16x16x4_f32(false, a, false, b2, (short)0, acc2, false, false);
            acc3 = __builtin_amdgcn_wmma_f32_16x16x4_f32(false, a, false, b3, (short)0, acc3, false, false);
        }
        if (wave == 0) __builtin_amdgcn_s_wait_tensorcnt(0);  // chunk c+1 done (overlapped DMA)
    }

    // C/D layout: VGPR r, lanes 0-15 -> (M=r, N=lane); lanes 16-31 -> (M=r+8, N=lane-16)
    float* wls = s_logits + wave * (TOKENS_PER_WAVE * ROW_STRIDE);
    #pragma unroll
    for (int r = 0; r < 8; ++r) {
        float* rowp = wls + (r + 8 * half) * ROW_STRIDE;
        rowp[ 0 + lmod] = acc0[r];
        rowp[16 + lmod] = acc1[r];
        rowp[32 + lmod] = acc2[r];
        rowp[48 + lmod] = acc3[r];
    }

    __syncthreads();

    if (half == 0) {  // lanes 0-15: one token each
        const int token = token_base + lane;
        float* row = wls + lane * ROW_STRIDE;

        // fp32 softmax over 64 experts (matches reference)
        float mx = -3.402823466e38f;
        #pragma unroll 8
        for (int e = 0; e < NUM_EXPERTS; ++e) mx = fmaxf(mx, row[e]);
        float sum = 0.0f;
        #pragma unroll 8
        for (int e = 0; e < NUM_EXPERTS; ++e) {
            float v = __expf(row[e] - mx);
            row[e] = v;
            sum += v;
        }
        const float inv = 1.0f / sum;
        #pragma unroll 8
        for (int e = 0; e < NUM_EXPERTS; ++e) row[e] *= inv;

        // biased top-8 (strict > keeps lowest index on ties, like jax top_k)
        int   sel_idx[TOP_K];
        float sel_scr[TOP_K];
        float ssum = 0.0f;
        #pragma unroll
        for (int j = 0; j < TOP_K; ++j) {
            float best = -3.402823466e38f;
            int   bi   = 0;
            #pragma unroll 8
            for (int e = 0; e < NUM_EXPERTS; ++e) {
                float v = row[e] + s_bias[e];
                if (v > best) { best = v; bi = e; }
            }
            sel_idx[j] = bi;
            sel_scr[j] = row[bi];   // pre-mask softmax score
            ssum += row[bi];
            row[bi] = -3.402823466e38f;
            atomicAdd(&s_counts[bi], 1.0f);
        }
        const float norm = 1.0f / (ssum + 1e-9f);
        #pragma unroll
        for (int j = 0; j < TOP_K; ++j) {
            out_idx[(size_t)token * TOP_K + j] = sel_idx[j];
            out_scr[(size_t)token * TOP_K + j] = sel_scr[j] * norm;
        }
    }

    __syncthreads();
    if (tid < NUM_EXPERTS) atomicAdd(&out_counts[tid], s_counts[tid]);
}

extern "C" void kernel_launch(void* const* d_in, const int* in_sizes, int n_in,
                              void* d_out, int out_size, void* d_ws, size_t ws_size,
                              hipStream_t stream) {
    const float* x    = (const float*)d_in[0];
    const float* gw   = (const float*)d_in[1];
    const float* bias = (const float*)d_in[2];

    // d_out: [ top_indices int32 (16384*8) | top_scores f32 (16384*8) | counts f32 (64) ]
    int*   out_idx = (int*)d_out;
    float* out_scr = (float*)d_out + (size_t)NUM_TOKENS * TOP_K;
    float* out_cnt = (float*)d_out + (size_t)2 * NUM_TOKENS * TOP_K;

    moe_zero_counts_kernel<<<1, NUM_EXPERTS, 0, stream>>>(out_cnt);

    const int blocks = NUM_TOKENS / TOKENS_PER_BLOCK;  // 256
    moe_router_kernel<<<blocks, WAVES_PER_BLOCK * 32, 0, stream>>>(
        x, gw, bias, out_idx, out_scr, out_cnt);
}